// DNC_3882650436261
// MI455X (gfx1250) — compile-verified
//
#include <hip/hip_runtime.h>
#include <math.h>

#define Bx   128
#define Mx   512
#define Wx   128
#define Rx   4
#define INx  1024
#define DELTAx 1e-6f

typedef float v2f __attribute__((ext_vector_type(2)));
typedef float v8f __attribute__((ext_vector_type(8)));

__device__ __forceinline__ float act_apply(float v, int act) {
  if (act == 0) return tanhf(v);                                   // tanh
  if (act == 1) return (v > 20.f) ? v : log1pf(__expf(v));         // softplus
  if (act == 2) return 1.f / (1.f + __expf(-v));                   // sigmoid
  return v;                                                        // identity
}

// ---------------------------------------------------------------------------
// Kernel 1: one 16x16 output tile of  out = act(x @ Wh^T + b),  K = 1024.
// One wave per block; f32 WMMA 16x16x4, 256 k-steps.
// A tile: x rows (batch), B tile: Wh rows (out cols), both f32.
// ---------------------------------------------------------------------------
__global__ __launch_bounds__(32)
void lin_wmma_kernel(const float* __restrict__ x, const float* __restrict__ Wh,
                     const float* __restrict__ bh, float* __restrict__ out,
                     int D, int act) {
  const int lane = threadIdx.x;
  const int half = lane >> 4;          // K half select
  const int lm   = lane & 15;
  const int ct   = blockIdx.x * 16;    // output-column tile
  const int bt   = blockIdx.y * 16;    // batch-row tile
  const int row  = bt + lm;            // A: M row
  const int col  = ct + lm;            // B: N col
  const bool cv  = (col < D);
  const float* xrow = x  + (size_t)row * INx;
  const float* wrow = Wh + (size_t)(cv ? col : 0) * INx;
  v8f c = {0.f,0.f,0.f,0.f,0.f,0.f,0.f,0.f};
  for (int k = 0; k < INx; k += 4) {
    const int ka = k + 2 * half;       // lane's K slots: ka, ka+1
    v2f a, bm;
    a.x = xrow[ka];
    a.y = xrow[ka + 1];
    float b0 = wrow[ka], b1 = wrow[ka + 1];
    bm.x = cv ? b0 : 0.f;
    bm.y = cv ? b1 : 0.f;
    c = __builtin_amdgcn_wmma_f32_16x16x4_f32(false, a, false, bm,
                                              (short)0, c, false, false);
  }
  if (cv) {
    const float bias = bh[col];
#pragma unroll
    for (int i = 0; i < 8; ++i) {
      const int r = bt + i + 8 * half; // C layout: VGPR i -> M = i (+8 hi half)
      out[(size_t)r * D + col] = act_apply(c[i] + bias, act);
    }
  }
}

// ---------------------------------------------------------------------------
// Kernel 2: usage update + retention psi (elementwise over B*M)
// ---------------------------------------------------------------------------
__global__ __launch_bounds__(256)
void usage_kernel(const float* __restrict__ uv, const float* __restrict__ wwts,
                  const float* __restrict__ fgA, const float* __restrict__ rwts,
                  float* __restrict__ usage) {
  const size_t i = (size_t)blockIdx.x * 256 + threadIdx.x;
  if (i >= (size_t)Bx * Mx) return;
  const int b = (int)(i / Mx), m = (int)(i % Mx);
  const float u = uv[i], w = wwts[i];
  float un = u + (1.f - u) * w;
  float psi = 1.f;
#pragma unroll
  for (int r = 0; r < Rx; ++r)
    psi *= 1.f - fgA[b * Rx + r] * rwts[((size_t)b * Rx + r) * Mx + m];
  usage[i] = un * psi;
}

// ---------------------------------------------------------------------------
// Kernel 3: write content weights: softmax_m( cos(memory, write_key)*strength )
// One workgroup (256 thr) per batch.
// ---------------------------------------------------------------------------
__global__ __launch_bounds__(256)
void wcw_kernel(const float* __restrict__ memp, const float* __restrict__ wk,
                const float* __restrict__ wstr, float* __restrict__ outw) {
  __shared__ __align__(16) float key[Wx];
  __shared__ float sc[Mx];
  __shared__ float red[256];
  const int b = blockIdx.x, t = threadIdx.x;
  if (t < Wx) key[t] = wk[(size_t)b * Wx + t];
  __syncthreads();
  float kn = 0.f;
  for (int w = 0; w < Wx; ++w) kn += key[w] * key[w];
  kn = sqrtf(kn) + DELTAx;
  const float st = wstr[b];
  for (int m = t; m < Mx; m += 256) {
    const float4* row = (const float4*)(memp + ((size_t)b * Mx + m) * Wx);
    float dot = 0.f, nn = 0.f;
    for (int w4 = 0; w4 < Wx / 4; ++w4) {
      const float4 v  = row[w4];
      const float4 kv = ((const float4*)key)[w4];
      dot += v.x * kv.x + v.y * kv.y + v.z * kv.z + v.w * kv.w;
      nn  += v.x * v.x  + v.y * v.y  + v.z * v.z  + v.w * v.w;
    }
    const float nrm = sqrtf(nn) + DELTAx;
    sc[m] = dot / ((float)Wx * kn * nrm + DELTAx) * st;
  }
  __syncthreads();
  float mx = -1e30f;
  for (int m = t; m < Mx; m += 256) mx = fmaxf(mx, sc[m]);
  red[t] = mx; __syncthreads();
  for (int s = 128; s > 0; s >>= 1) { if (t < s) red[t] = fmaxf(red[t], red[t + s]); __syncthreads(); }
  mx = red[0]; __syncthreads();
  float sum = 0.f;
  for (int m = t; m < Mx; m += 256) { const float e = __expf(sc[m] - mx); sc[m] = e; sum += e; }
  red[t] = sum; __syncthreads();
  for (int s = 128; s > 0; s >>= 1) { if (t < s) red[t] += red[t + s]; __syncthreads(); }
  sum = red[0]; __syncthreads();
  const float inv = 1.f / sum;
  for (int m = t; m < Mx; m += 256) outw[(size_t)b * Mx + m] = sc[m] * inv;
}

// ---------------------------------------------------------------------------
// Kernel 4: allocation weighting (sort by usage, exclusive cumprod, scatter)
// LDS bitonic sort of 512 (value,index) pairs; one workgroup per batch.
// ---------------------------------------------------------------------------
__global__ __launch_bounds__(256)
void alloc_kernel(const float* __restrict__ usage, float* __restrict__ alloc) {
  __shared__ float val[Mx];
  __shared__ int   idx[Mx];
  __shared__ float pr[Mx];
  const int b = blockIdx.x, t = threadIdx.x;
  for (int m = t; m < Mx; m += 256) {
    val[m] = DELTAx + (1.f - DELTAx) * usage[(size_t)b * Mx + m];
    idx[m] = m;
  }
  __syncthreads();
  for (int k = 2; k <= Mx; k <<= 1) {
    for (int j = k >> 1; j > 0; j >>= 1) {
      for (int i = t; i < Mx; i += 256) {
        const int ixj = i ^ j;
        if (ixj > i) {
          const bool up = ((i & k) == 0);
          const float a = val[i], bb = val[ixj];
          const bool sw = up ? (a > bb) : (a < bb);
          if (sw) {
            val[i] = bb; val[ixj] = a;
            const int tmp = idx[i]; idx[i] = idx[ixj]; idx[ixj] = tmp;
          }
        }
      }
      __syncthreads();
    }
  }
  if (t == 0) {           // exclusive cumulative product of sorted usage
    float p = 1.f;
    for (int m = 0; m < Mx; ++m) { pr[m] = p; p *= val[m]; }
  }
  __syncthreads();
  for (int m = t; m < Mx; m += 256)
    alloc[(size_t)b * Mx + idx[m]] = (1.f - val[m]) * pr[m];
}

// ---------------------------------------------------------------------------
// Kernel 5: ww = wg*(ag*alloc + (1-ag)*wcw)
// ---------------------------------------------------------------------------
__global__ __launch_bounds__(256)
void ww_kernel(const float* __restrict__ alloc, const float* __restrict__ wcw,
               const float* __restrict__ agA, const float* __restrict__ wgA,
               float* __restrict__ ww) {
  const size_t i = (size_t)blockIdx.x * 256 + threadIdx.x;
  if (i >= (size_t)Bx * Mx) return;
  const int b = (int)(i / Mx);
  const float a = agA[b], g = wgA[b];
  ww[i] = g * (a * alloc[i] + (1.f - a) * wcw[i]);
}

// ---------------------------------------------------------------------------
// Kernel 6: mem_new = memory*(1 - ww*ev) + ww*wv   (float4 over B*M*W)
// ---------------------------------------------------------------------------
__global__ __launch_bounds__(256)
void memnew_kernel(const float* __restrict__ memp, const float* __restrict__ ww,
                   const float* __restrict__ ev, const float* __restrict__ wv,
                   float* __restrict__ memn) {
  const size_t i4 = (size_t)blockIdx.x * 256 + threadIdx.x;
  const size_t tot = (size_t)Bx * Mx * (Wx / 4);
  if (i4 >= tot) return;
  const int    w4 = (int)(i4 % (Wx / 4));
  const size_t bm = i4 / (Wx / 4);
  const int    b  = (int)(bm / Mx);
  const float wwv = ww[bm];
  const float4 mv = ((const float4*)memp)[i4];
  const float4 e  = ((const float4*)ev)[(size_t)b * (Wx / 4) + w4];
  const float4 v  = ((const float4*)wv)[(size_t)b * (Wx / 4) + w4];
  float4 o;
  o.x = mv.x * (1.f - wwv * e.x) + wwv * v.x;
  o.y = mv.y * (1.f - wwv * e.y) + wwv * v.y;
  o.z = mv.z * (1.f - wwv * e.z) + wwv * v.z;
  o.w = mv.w * (1.f - wwv * e.w) + wwv * v.w;
  ((float4*)memn)[i4] = o;
}

// ---------------------------------------------------------------------------
// Kernel 7: read content weights on mem_new (4 keys), workgroup per batch
// ---------------------------------------------------------------------------
__global__ __launch_bounds__(256)
void cw_kernel(const float* __restrict__ memn, const float* __restrict__ rk,
               const float* __restrict__ rs, float* __restrict__ cw) {
  __shared__ __align__(16) float keys[Rx * Wx];
  __shared__ float sc[Rx * Mx];
  __shared__ float red[256];
  const int b = blockIdx.x, t = threadIdx.x;
  for (int i = t; i < Rx * Wx; i += 256) keys[i] = rk[(size_t)b * Rx * Wx + i];
  __syncthreads();
  float kn[Rx];
#pragma unroll
  for (int r = 0; r < Rx; ++r) {
    float s = 0.f;
    for (int w = 0; w < Wx; ++w) { const float kv = keys[r * Wx + w]; s += kv * kv; }
    kn[r] = sqrtf(s) + DELTAx;
  }
  for (int m = t; m < Mx; m += 256) {
    const float4* row = (const float4*)(memn + ((size_t)b * Mx + m) * Wx);
    float nn = 0.f;
    float dot[Rx] = {0.f, 0.f, 0.f, 0.f};
    for (int w4 = 0; w4 < Wx / 4; ++w4) {
      const float4 v = row[w4];
      nn += v.x * v.x + v.y * v.y + v.z * v.z + v.w * v.w;
#pragma unroll
      for (int r = 0; r < Rx; ++r) {
        const float4 kv = ((const float4*)keys)[r * (Wx / 4) + w4];
        dot[r] += v.x * kv.x + v.y * kv.y + v.z * kv.z + v.w * kv.w;
      }
    }
    const float nrm = sqrtf(nn) + DELTAx;
#pragma unroll
    for (int r = 0; r < Rx; ++r)
      sc[r * Mx + m] = dot[r] / ((float)Wx * kn[r] * nrm + DELTAx) * rs[b * Rx + r];
  }
  __syncthreads();
  for (int r = 0; r < Rx; ++r) {
    float mx = -1e30f;
    for (int m = t; m < Mx; m += 256) mx = fmaxf(mx, sc[r * Mx + m]);
    red[t] = mx; __syncthreads();
    for (int s = 128; s > 0; s >>= 1) { if (t < s) red[t] = fmaxf(red[t], red[t + s]); __syncthreads(); }
    mx = red[0]; __syncthreads();
    float sum = 0.f;
    for (int m = t; m < Mx; m += 256) { const float e = __expf(sc[r * Mx + m] - mx); sc[r * Mx + m] = e; sum += e; }
    red[t] = sum; __syncthreads();
    for (int s = 128; s > 0; s >>= 1) { if (t < s) red[t] += red[t + s]; __syncthreads(); }
    sum = red[0]; __syncthreads();
    const float inv = 1.f / sum;
    for (int m = t; m < Mx; m += 256) cw[((size_t)b * Rx + r) * Mx + m] = sc[r * Mx + m] * inv;
  }
}

// ---------------------------------------------------------------------------
// Kernel 8: fused link update + fwd/bwd. Streams link_matrix (134 MB) ONCE,
// never materializing the updated link. Workgroup (8 waves) per batch.
//   L[n][c] = (c==n) ? 0 : (1 - ww[n] - ww[c])*LM[n][c] + ww[n]*prec[c]
//   fwd[r][n] = sum_c L[n][c]*rw[r][c]   (wave shfl-xor reduction)
//   bwd[r][c] += rw[r][n]*L[n][c]        (per-lane regs, deterministic flush)
// ---------------------------------------------------------------------------
__global__ __launch_bounds__(256)
void link_kernel(const float* __restrict__ LM, const float* __restrict__ prec,
                 const float* __restrict__ rwts, const float* __restrict__ ww,
                 float* __restrict__ fwd, float* __restrict__ bwd) {
  __shared__ float rwS[Rx * Mx];
  __shared__ float precS[Mx];
  __shared__ float wwS[Mx];
  __shared__ float fwdS[Rx * Mx];
  __shared__ float bwdS[Rx * Mx];
  const int b = blockIdx.x, t = threadIdx.x;
  const int wave = t >> 5, lane = t & 31;
  for (int i = t; i < Rx * Mx; i += 256) { rwS[i] = rwts[(size_t)b * Rx * Mx + i]; bwdS[i] = 0.f; }
  for (int i = t; i < Mx; i += 256) { precS[i] = prec[(size_t)b * Mx + i]; wwS[i] = ww[(size_t)b * Mx + i]; }
  __syncthreads();

  float bacc[Rx][4][4];
#pragma unroll
  for (int r = 0; r < Rx; ++r)
#pragma unroll
    for (int j = 0; j < 4; ++j)
#pragma unroll
      for (int q = 0; q < 4; ++q) bacc[r][j][q] = 0.f;

  for (int n = wave; n < Mx; n += 8) {
    const float wwn = wwS[n];
    const float rn[Rx] = { rwS[0 * Mx + n], rwS[1 * Mx + n], rwS[2 * Mx + n], rwS[3 * Mx + n] };
    float fr[Rx] = {0.f, 0.f, 0.f, 0.f};
    const float4* lrow = (const float4*)(LM + ((size_t)b * Mx + n) * Mx);
#pragma unroll
    for (int j = 0; j < 4; ++j) {
      const int c0 = j * 128 + lane * 4;
      const float4 l4 = lrow[j * 32 + lane];
      const float lv[4] = { l4.x, l4.y, l4.z, l4.w };
#pragma unroll
      for (int q = 0; q < 4; ++q) {
        const int c = c0 + q;
        const float Lnc = (c == n) ? 0.f
                        : ((1.f - wwn - wwS[c]) * lv[q] + wwn * precS[c]);
#pragma unroll
        for (int r = 0; r < Rx; ++r) {
          fr[r] += Lnc * rwS[r * Mx + c];
          bacc[r][j][q] += rn[r] * Lnc;
        }
      }
    }
#pragma unroll
    for (int r = 0; r < Rx; ++r) {
      float v = fr[r];
      for (int off = 16; off > 0; off >>= 1) v += __shfl_xor(v, off, 32);
      if (lane == 0) fwdS[r * Mx + n] = v;
    }
  }
  // deterministic, serialized flush of per-lane bwd accumulators into LDS
  for (int wv = 0; wv < 8; ++wv) {
    if (wave == wv) {
#pragma unroll
      for (int r = 0; r < Rx; ++r)
#pragma unroll
        for (int j = 0; j < 4; ++j)
#pragma unroll
          for (int q = 0; q < 4; ++q)
            bwdS[r * Mx + j * 128 + lane * 4 + q] += bacc[r][j][q];
    }
    __syncthreads();
  }
  for (int i = t; i < Rx * Mx; i += 256) {
    fwd[(size_t)b * Rx * Mx + i] = fwdS[i];
    bwd[(size_t)b * Rx * Mx + i] = bwdS[i];
  }
}

// ---------------------------------------------------------------------------
// Kernel 9: rw_new = softmax3(rm)·{bwd, fwd, cw}
// ---------------------------------------------------------------------------
__global__ __launch_bounds__(256)
void rwn_kernel(const float* __restrict__ rmraw, const float* __restrict__ fwd,
                const float* __restrict__ bwd, const float* __restrict__ cw,
                float* __restrict__ rwn) {
  const size_t i = (size_t)blockIdx.x * 256 + threadIdx.x;
  if (i >= (size_t)Bx * Rx * Mx) return;
  const size_t br = i / Mx;
  const int b = (int)(br / Rx), r = (int)(br % Rx);
  const float* e = rmraw + (size_t)b * (3 * Rx) + r * 3;
  const float m  = fmaxf(e[0], fmaxf(e[1], e[2]));
  const float e0 = __expf(e[0] - m), e1 = __expf(e[1] - m), e2 = __expf(e[2] - m);
  const float inv = 1.f / (e0 + e1 + e2);
  rwn[i] = (e0 * bwd[i] + e1 * fwd[i] + e2 * cw[i]) * inv;
}

// ---------------------------------------------------------------------------
// Kernel 10: read_vectors = rw_new[4xM] @ mem_new[MxW] per batch, via f32 WMMA.
// Rows padded 4->16; grid (W/16 tiles, B); one wave per block; 128 k-steps.
// ---------------------------------------------------------------------------
__global__ __launch_bounds__(32)
void readvec_wmma_kernel(const float* __restrict__ rwn,
                         const float* __restrict__ memn,
                         float* __restrict__ out) {
  const int b    = blockIdx.y;
  const int wt   = blockIdx.x * 16;
  const int lane = threadIdx.x;
  const int half = lane >> 4;
  const int lm   = lane & 15;
  const int wcol = wt + lm;
  const bool av  = (lm < Rx);
  const float* arow = rwn + ((size_t)b * Rx + (av ? lm : 0)) * Mx;
  v8f c = {0.f,0.f,0.f,0.f,0.f,0.f,0.f,0.f};
  for (int k = 0; k < Mx; k += 4) {
    const int ka = k + 2 * half;
    v2f a, bm;
    float a0 = arow[ka], a1 = arow[ka + 1];
    a.x = av ? a0 : 0.f;
    a.y = av ? a1 : 0.f;
    bm.x = memn[((size_t)b * Mx + ka)     * Wx + wcol];
    bm.y = memn[((size_t)b * Mx + ka + 1) * Wx + wcol];
    c = __builtin_amdgcn_wmma_f32_16x16x4_f32(false, a, false, bm,
                                              (short)0, c, false, false);
  }
  if (half == 0) {
#pragma unroll
    for (int i = 0; i < Rx; ++i)
      out[((size_t)b * Rx + i) * Wx + wcol] = c[i];
  }
}

// ---------------------------------------------------------------------------
extern "C" void kernel_launch(void* const* d_in, const int* in_sizes, int n_in,
                              void* d_out, int out_size, void* d_ws, size_t ws_size,
                              hipStream_t stream) {
  (void)in_sizes; (void)n_in; (void)out_size; (void)ws_size;
  const float* x    = (const float*)d_in[0];
  const float* memp = (const float*)d_in[1];
  const float* LM   = (const float*)d_in[2];
  const float* prec = (const float*)d_in[3];
  const float* rwts = (const float*)d_in[4];
  const float* wwts = (const float*)d_in[5];
  const float* uv   = (const float*)d_in[6];
  const float* Wrk = (const float*)d_in[7];  const float* brk = (const float*)d_in[8];
  const float* Wrs = (const float*)d_in[9];  const float* brs = (const float*)d_in[10];
  const float* Wwk = (const float*)d_in[11]; const float* bwk = (const float*)d_in[12];
  const float* Wws = (const float*)d_in[13]; const float* bws = (const float*)d_in[14];
  const float* Wev = (const float*)d_in[15]; const float* bev = (const float*)d_in[16];
  const float* Wwv = (const float*)d_in[17]; const float* bwv = (const float*)d_in[18];
  const float* Wfg = (const float*)d_in[19]; const float* bfg = (const float*)d_in[20];
  const float* Wag = (const float*)d_in[21]; const float* bag = (const float*)d_in[22];
  const float* Wwg = (const float*)d_in[23]; const float* bwg = (const float*)d_in[24];
  const float* Wrm = (const float*)d_in[25]; const float* brm = (const float*)d_in[26];
  float* out = (float*)d_out;

  float* ws = (float*)d_ws;
  size_t off = 0;
  auto take = [&](size_t n) {
    float* p = ws + off;
    off += (n + 255) & ~(size_t)255;   // keep 1KB-multiple offsets (float4 safe)
    return p;
  };
  float* rkA  = take((size_t)Bx * Rx * Wx);
  float* rsA  = take((size_t)Bx * Rx);
  float* wkA  = take((size_t)Bx * Wx);
  float* wsA  = take((size_t)Bx);
  float* evA  = take((size_t)Bx * Wx);
  float* wvA  = take((size_t)Bx * Wx);
  float* fgA  = take((size_t)Bx * Rx);
  float* agA  = take((size_t)Bx);
  float* wgA  = take((size_t)Bx);
  float* rmA  = take((size_t)Bx * 3 * Rx);
  float* usg  = take((size_t)Bx * Mx);
  float* wcw  = take((size_t)Bx * Mx);
  float* alc  = take((size_t)Bx * Mx);
  float* wwb  = take((size_t)Bx * Mx);
  float* memn = take((size_t)Bx * Mx * Wx);
  float* cwb  = take((size_t)Bx * Rx * Mx);
  float* fwdb = take((size_t)Bx * Rx * Mx);
  float* bwdb = take((size_t)Bx * Rx * Mx);
  float* rwnb = take((size_t)Bx * Rx * Mx);

  auto lin = [&](const float* W, const float* bvec, float* o, int D, int act) {
    dim3 grid((D + 15) / 16, Bx / 16);
    lin_wmma_kernel<<<grid, dim3(32), 0, stream>>>(x, W, bvec, o, D, act);
  };
  lin(Wrk, brk, rkA, Rx * Wx, 0);   // tanh
  lin(Wrs, brs, rsA, Rx,      1);   // softplus
  lin(Wwk, bwk, wkA, Wx,      0);   // tanh
  lin(Wws, bws, wsA, 1,       1);   // softplus
  lin(Wev, bev, evA, Wx,      2);   // sigmoid
  lin(Wwv, bwv, wvA, Wx,      0);   // tanh
  lin(Wfg, bfg, fgA, Rx,      2);   // sigmoid
  lin(Wag, bag, agA, 1,       2);   // sigmoid
  lin(Wwg, bwg, wgA, 1,       2);   // sigmoid
  lin(Wrm, brm, rmA, 3 * Rx,  3);   // raw (softmax later)

  usage_kernel<<<(Bx * Mx + 255) / 256, 256, 0, stream>>>(uv, wwts, fgA, rwts, usg);
  wcw_kernel<<<Bx, 256, 0, stream>>>(memp, wkA, wsA, wcw);
  alloc_kernel<<<Bx, 256, 0, stream>>>(usg, alc);
  ww_kernel<<<(Bx * Mx + 255) / 256, 256, 0, stream>>>(alc, wcw, agA, wgA, wwb);
  memnew_kernel<<<((Bx * Mx * Wx / 4) + 255) / 256, 256, 0, stream>>>(memp, wwb, evA, wvA, memn);
  cw_kernel<<<Bx, 256, 0, stream>>>(memn, rkA, rsA, cwb);
  link_kernel<<<Bx, 256, 0, stream>>>(LM, prec, rwts, wwb, fwdb, bwdb);
  rwn_kernel<<<(Bx * Rx * Mx + 255) / 256, 256, 0, stream>>>(rmA, fwdb, bwdb, cwb, rwnb);
  readvec_wmma_kernel<<<dim3(Wx / 16, Bx), dim3(32), 0, stream>>>(rwnb, memn, out);
}